// HeteroAttentionGNN_34832184771029
// MI455X (gfx1250) — compile-verified
//
#include <hip/hip_runtime.h>
#include <math.h>

typedef float v2f __attribute__((ext_vector_type(2)));
typedef float v8f __attribute__((ext_vector_type(8)));

// ---------- helpers ----------
__device__ __forceinline__ unsigned f2ord(float f) {
  unsigned b = __float_as_uint(f);
  return (b & 0x80000000u) ? ~b : (b | 0x80000000u);   // monotone float->uint map
}
__device__ __forceinline__ float ord2f(unsigned k) {
  unsigned b = (k & 0x80000000u) ? (k & 0x7FFFFFFFu) : ~k;
  return __uint_as_float(b);
}
__device__ __forceinline__ float lrelu(float v) { return v > 0.f ? v : 0.2f * v; }

// ---------- fp32 WMMA GEMM: Y[N x 32] = X[N x K] @ W[K x 32] (+ optional bias) ----------
// One wave computes a 16-row x 32-col tile using V_WMMA_F32_16X16X4_F32.
// A layout (16x4 f32): lanes 0-15 hold M=0..15 {K=k0,k0+1}, lanes 16-31 hold {K=k0+2,k0+3}.
// B/C layout: rows striped across lanes within a VGPR (C vgpr r: lanes0-15 M=r, lanes16-31 M=r+8).
// W (K x 32, K <= 128) is staged in LDS once per block; 8 waves share it.
__global__ void k_gemm_wmma(const float* __restrict__ X, const float* __restrict__ W,
                            const float* __restrict__ bias, float* __restrict__ Y,
                            int N, int K) {
  __shared__ float sW[128 * 32];                 // max K = 128 -> 16 KB
  for (int i = threadIdx.x; i < K * 32; i += blockDim.x) sW[i] = W[i];
  __syncthreads();

  const int lane = threadIdx.x & 31;
  const int wave = threadIdx.x >> 5;
  const int tile = blockIdx.x * (blockDim.x >> 5) + wave;
  const int row0 = tile * 16;
  if (row0 >= N) return;                         // uniform per wave: EXEC stays all-ones
  const int hi  = lane >> 4;                     // half-wave select
  const int l15 = lane & 15;

  // A row for this lane: clamp instead of predicating the load (wrong rows never stored)
  const int arow = min(row0 + l15, N - 1);
  const float* __restrict__ Arow = X + (size_t)arow * K + (hi << 1);

  v8f c0 = {};
  v8f c1 = {};
  for (int k0 = 0; k0 < K; k0 += 4) {
    const int kk = k0 + (hi << 1);               // k0 or k0+2
    v2f a;
    a.x = Arow[k0];
    a.y = Arow[k0 + 1];
    v2f b0, b1;
    b0.x = sW[kk * 32 + l15];        b0.y = sW[(kk + 1) * 32 + l15];
    b1.x = sW[kk * 32 + 16 + l15];   b1.y = sW[(kk + 1) * 32 + 16 + l15];
    c0 = __builtin_amdgcn_wmma_f32_16x16x4_f32(false, a, false, b0, (short)0, c0, false, false);
    c1 = __builtin_amdgcn_wmma_f32_16x16x4_f32(false, a, false, b1, (short)0, c1, false, false);
  }
  const int mOff = hi * 8;
#pragma unroll
  for (int r = 0; r < 8; ++r) {
    const int row = row0 + mOff + r;
    if (row < N) {
      float v0 = c0[r], v1 = c1[r];
      if (bias) { v0 += bias[l15]; v1 += bias[16 + l15]; }
      Y[(size_t)row * 32 + l15]      = v0;
      Y[(size_t)row * 32 + 16 + l15] = v1;
    }
  }
}

// ---------- layer-1 attention coefficients: al_s/al_d [N,2] ----------
__global__ void k_alpha1(const float* __restrict__ hs, const float* __restrict__ hd,
                         const float* __restrict__ a1s, const float* __restrict__ a1d,
                         float* __restrict__ al_s, float* __restrict__ al_d, int N) {
  int tid = blockIdx.x * blockDim.x + threadIdx.x;
  if (tid >= N * 2) return;
  int n = tid >> 1, h = tid & 1;
  float ss = 0.f, sd = 0.f;
#pragma unroll
  for (int c = 0; c < 16; ++c) {
    ss += hs[(size_t)n * 32 + h * 16 + c] * a1s[h * 16 + c];
    sd += hd[(size_t)n * 32 + h * 16 + c] * a1d[h * 16 + c];
  }
  al_s[tid] = ss;
  al_d[tid] = sd;
}

// ---------- per-layer softmax state init ----------
__global__ void k_init(unsigned* __restrict__ maxb, float* __restrict__ sumb,
                       float* __restrict__ agg, int nh, int nfeat) {
  int tid = blockIdx.x * blockDim.x + threadIdx.x;
  if (tid < nh) { maxb[tid] = 0u; sumb[tid] = 0.f; }   // 0u < key of any finite float
  if (tid < nfeat) agg[tid] = 0.f;
}

// ---------- edge logits ----------
__global__ void k_edge_logits_gat1(const float* __restrict__ al_s, const float* __restrict__ al_d,
                                   const int* __restrict__ src, const int* __restrict__ dst,
                                   int E, unsigned* __restrict__ maxb, float* __restrict__ edgeL) {
  int tid = blockIdx.x * blockDim.x + threadIdx.x;
  if (tid >= E * 2) return;
  int e = tid >> 1, h = tid & 1;
  float v = lrelu(al_s[src[e] * 2 + h] + al_d[dst[e] * 2 + h]);
  edgeL[tid] = v;
  atomicMax(&maxb[dst[e] * 2 + h], f2ord(v));
}

__global__ void k_edge_logits_v2(const float* __restrict__ hl, const float* __restrict__ hr,
                                 const float* __restrict__ a2,
                                 const int* __restrict__ src, const int* __restrict__ dst,
                                 int E, unsigned* __restrict__ maxb, float* __restrict__ edgeL) {
  int tid = blockIdx.x * blockDim.x + threadIdx.x;
  if (tid >= E * 2) return;
  int e = tid >> 1, h = tid & 1;
  int s = src[e], d = dst[e];
  float acc = 0.f;
#pragma unroll
  for (int c = 0; c < 16; ++c)
    acc += lrelu(hl[(size_t)s * 32 + h * 16 + c] + hr[(size_t)d * 32 + h * 16 + c]) * a2[h * 16 + c];
  edgeL[tid] = acc;
  atomicMax(&maxb[d * 2 + h], f2ord(acc));
}

__global__ void k_edge_logits_tr(const float* __restrict__ q, const float* __restrict__ kk,
                                 const int* __restrict__ src, const int* __restrict__ dst,
                                 int E, unsigned* __restrict__ maxb, float* __restrict__ edgeL) {
  int e = blockIdx.x * blockDim.x + threadIdx.x;
  if (e >= E) return;
  int s = src[e], d = dst[e];
  float acc = 0.f;
#pragma unroll
  for (int c = 0; c < 32; ++c) acc += q[(size_t)d * 32 + c] * kk[(size_t)s * 32 + c];
  acc *= 0.17677669529663687f;               // 1/sqrt(32)
  edgeL[e] = acc;
  atomicMax(&maxb[d], f2ord(acc));
}

// ---------- exp + segment sum ----------
__global__ void k_edge_exp(float* __restrict__ edgeL, const int* __restrict__ dst,
                           const unsigned* __restrict__ maxb, float* __restrict__ sumb,
                           int total, int hshift) {
  int tid = blockIdx.x * blockDim.x + threadIdx.x;
  if (tid >= total) return;
  int e = tid >> hshift, h = tid & ((1 << hshift) - 1);
  int seg = dst[e] * (1 << hshift) + h;
  float v = __expf(edgeL[tid] - ord2f(maxb[seg]));
  edgeL[tid] = v;
  atomicAdd(&sumb[seg], v);
}

// ---------- normalize + weighted scatter-add of source features ----------
__global__ void k_edge_scatter(const float* __restrict__ edgeL, const float* __restrict__ feat,
                               const int* __restrict__ src, const int* __restrict__ dst,
                               const float* __restrict__ sumb, float* __restrict__ agg,
                               long long total, int cshift, int hshift) {
  long long tid = (long long)blockIdx.x * blockDim.x + threadIdx.x;
  if (tid >= total) return;
  int C = 1 << cshift, H = 1 << hshift;
  int c  = (int)(tid & (C - 1));
  long long eh = tid >> cshift;
  int h  = (int)(eh & (H - 1));
  long long e = eh >> hshift;
  int s = src[e], d = dst[e];
  float w = edgeL[eh] / (sumb[d * H + h] + 1e-16f);
  atomicAdd(&agg[(size_t)d * (H * C) + h * C + c], feat[(size_t)s * (H * C) + h * C + c] * w);
}

// ---------- node updates ----------
__global__ void k_bias_relu(const float* __restrict__ agg, const float* __restrict__ bias,
                            float* __restrict__ out, int total) {
  int tid = blockIdx.x * blockDim.x + threadIdx.x;
  if (tid >= total) return;
  out[tid] = fmaxf(agg[tid] + bias[tid & 31], 0.f);
}

__global__ void k_add(const float* __restrict__ a, const float* __restrict__ b,
                      float* __restrict__ out, int total) {
  int tid = blockIdx.x * blockDim.x + threadIdx.x;
  if (tid >= total) return;
  out[tid] = a[tid] + b[tid];
}

__global__ void k_head(const float* __restrict__ x3, const float* __restrict__ Wlin,
                       const float* __restrict__ blin, float* __restrict__ out, int N) {
  int n = blockIdx.x * blockDim.x + threadIdx.x;
  if (n >= N) return;
  float s = blin[0];
#pragma unroll
  for (int c = 0; c < 32; ++c) s += x3[(size_t)n * 32 + c] * Wlin[c];
  out[n] = s;
}

static inline int nblk(long long n, int b) { return (int)((n + b - 1) / b); }

extern "C" void kernel_launch(void* const* d_in, const int* in_sizes, int n_in,
                              void* d_out, int out_size, void* d_ws, size_t ws_size,
                              hipStream_t stream) {
  const float* x    = (const float*)d_in[0];
  const int*   ei   = (const int*)d_in[1];
  const float* W1s  = (const float*)d_in[2];
  const float* W1d  = (const float*)d_in[3];
  const float* a1s  = (const float*)d_in[4];
  const float* a1d  = (const float*)d_in[5];
  const float* b1   = (const float*)d_in[6];
  const float* W2l  = (const float*)d_in[7];
  const float* b2l  = (const float*)d_in[8];
  const float* W2r  = (const float*)d_in[9];
  const float* b2r  = (const float*)d_in[10];
  const float* a2   = (const float*)d_in[11];
  const float* b2   = (const float*)d_in[12];
  const float* Wq   = (const float*)d_in[13];
  const float* bq   = (const float*)d_in[14];
  const float* Wk   = (const float*)d_in[15];
  const float* bk   = (const float*)d_in[16];
  const float* Wv   = (const float*)d_in[17];
  const float* bv   = (const float*)d_in[18];
  const float* Wsk  = (const float*)d_in[19];
  const float* bsk  = (const float*)d_in[20];
  const float* Wlin = (const float*)d_in[21];
  const float* blin = (const float*)d_in[22];

  const int N = in_sizes[0] / 128;
  const int E = in_sizes[1] / 2;
  const int* src = ei;
  const int* dst = ei + E;

  // workspace carve-out (all fp32 unless noted)
  float* ws   = (float*)d_ws;
  float* bufA = ws;                              // N*32  (hs / hl / q / x3)
  float* bufB = bufA + (size_t)N * 32;           // N*32  (hd / hr / k)
  float* bufC = bufB + (size_t)N * 32;           // N*32  (x1 / x2)
  float* bufD = bufC + (size_t)N * 32;           // N*32  (agg)
  float* bufE = bufD + (size_t)N * 32;           // N*32  (v)
  float* bufF = bufE + (size_t)N * 32;           // N*32  (skip)
  float* al_s = bufF + (size_t)N * 32;           // N*2
  float* al_d = al_s + (size_t)N * 2;            // N*2
  unsigned* maxb = (unsigned*)(al_d + (size_t)N * 2); // N*2 (ordered-uint max)
  float* sumb = (float*)(maxb + (size_t)N * 2);  // N*2
  float* edgeL = sumb + (size_t)N * 2;           // E*2

  const int B = 256;
  const int gemmBlocks = nblk((N + 15) / 16, 8); // 8 waves per block, 16 rows per wave

  // ======== Layer 1: GATConv (heads=2, C=16) ========
  k_gemm_wmma<<<gemmBlocks, B, 0, stream>>>(x, W1s, nullptr, bufA, N, 128);
  k_gemm_wmma<<<gemmBlocks, B, 0, stream>>>(x, W1d, nullptr, bufB, N, 128);
  k_alpha1<<<nblk((long long)N * 2, B), B, 0, stream>>>(bufA, bufB, a1s, a1d, al_s, al_d, N);
  k_init<<<nblk((long long)N * 32, B), B, 0, stream>>>(maxb, sumb, bufD, N * 2, N * 32);
  k_edge_logits_gat1<<<nblk((long long)E * 2, B), B, 0, stream>>>(al_s, al_d, src, dst, E, maxb, edgeL);
  k_edge_exp<<<nblk((long long)E * 2, B), B, 0, stream>>>(edgeL, dst, maxb, sumb, E * 2, 1);
  k_edge_scatter<<<nblk((long long)E * 32, B), B, 0, stream>>>(edgeL, bufA, src, dst, sumb, bufD,
                                                               (long long)E * 32, 4, 1);
  k_bias_relu<<<nblk((long long)N * 32, B), B, 0, stream>>>(bufD, b1, bufC, N * 32);

  // ======== Layer 2: GATv2Conv (heads=2, C=16) ========
  k_gemm_wmma<<<gemmBlocks, B, 0, stream>>>(bufC, W2l, b2l, bufA, N, 32);
  k_gemm_wmma<<<gemmBlocks, B, 0, stream>>>(bufC, W2r, b2r, bufB, N, 32);
  k_init<<<nblk((long long)N * 32, B), B, 0, stream>>>(maxb, sumb, bufD, N * 2, N * 32);
  k_edge_logits_v2<<<nblk((long long)E * 2, B), B, 0, stream>>>(bufA, bufB, a2, src, dst, E, maxb, edgeL);
  k_edge_exp<<<nblk((long long)E * 2, B), B, 0, stream>>>(edgeL, dst, maxb, sumb, E * 2, 1);
  k_edge_scatter<<<nblk((long long)E * 32, B), B, 0, stream>>>(edgeL, bufA, src, dst, sumb, bufD,
                                                               (long long)E * 32, 4, 1);
  k_bias_relu<<<nblk((long long)N * 32, B), B, 0, stream>>>(bufD, b2, bufC, N * 32);

  // ======== Layer 3: TransformerConv (heads=1, C=32) ========
  k_gemm_wmma<<<gemmBlocks, B, 0, stream>>>(bufC, Wq, bq, bufA, N, 32);
  k_gemm_wmma<<<gemmBlocks, B, 0, stream>>>(bufC, Wk, bk, bufB, N, 32);
  k_gemm_wmma<<<gemmBlocks, B, 0, stream>>>(bufC, Wv, bv, bufE, N, 32);
  k_gemm_wmma<<<gemmBlocks, B, 0, stream>>>(bufC, Wsk, bsk, bufF, N, 32);
  k_init<<<nblk((long long)N * 32, B), B, 0, stream>>>(maxb, sumb, bufD, N, N * 32);
  k_edge_logits_tr<<<nblk((long long)E, B), B, 0, stream>>>(bufA, bufB, src, dst, E, maxb, edgeL);
  k_edge_exp<<<nblk((long long)E, B), B, 0, stream>>>(edgeL, dst, maxb, sumb, E, 0);
  k_edge_scatter<<<nblk((long long)E * 32, B), B, 0, stream>>>(edgeL, bufE, src, dst, sumb, bufD,
                                                               (long long)E * 32, 5, 0);
  k_add<<<nblk((long long)N * 32, B), B, 0, stream>>>(bufD, bufF, bufA, N * 32);

  // ======== Head: Linear(32, 1) ========
  k_head<<<nblk((long long)N, B), B, 0, stream>>>(bufA, Wlin, blin, (float*)d_out, N);
}